// ResidualDeformableBlock_76647986365210
// MI455X (gfx1250) — compile-verified
//
#include <hip/hip_runtime.h>
#include <math.h>

// Problem constants (match reference)
#define B_   4
#define C_   64
#define H_   96
#define W_   96
#define HW_  (H_ * W_)        // 9216
#define KK_  9
#define KTOT (C_ * KK_)       // 576
#define NPIX (B_ * HW_)       // 36864
#define EPS_ 1e-5f

typedef float v2f __attribute__((ext_vector_type(2)));
typedef float v8f __attribute__((ext_vector_type(8)));

__device__ __forceinline__ v8f wmma4(v2f a, v2f b, v8f c) {
  // D = A(16x4,f32) * B(4x16,f32) + C(16x16,f32)  -> v_wmma_f32_16x16x4_f32
  return __builtin_amdgcn_wmma_f32_16x16x4_f32(false, a, false, b, (short)0, c,
                                               false, false);
}

__device__ __forceinline__ v8f zero8() {
  v8f z = {0.f, 0.f, 0.f, 0.f, 0.f, 0.f, 0.f, 0.f};
  return z;
}

__device__ __forceinline__ float sigmoidf_(float t) {
  return 1.f / (1.f + __expf(-t));
}

// ---------------------------------------------------------------------------
// Kernel 1: offset(18ch)+mask(9ch) 3x3 conv as WMMA GEMM (N=32 padded).
// K dimension iterated k-major (kidx' = k*64 + c) so the tap offset/validity
// is loop-invariant in the inner 16-step c-loop; gathers are unconditional
// (clamped address + cndmask zero) to avoid EXEC-divergent branches.
// Epilogue: per-(pixel,k) bilinear sampling setup: 4 indices + 4 folded
// weights (corner weight * validity * 2*sigmoid(mask)).
// One wave per 16-pixel tile; 4 waves per block.
// ---------------------------------------------------------------------------
__global__ __launch_bounds__(128)
void k_sample_setup(const float* __restrict__ src,    // (B,C,H,W)
                    const float* __restrict__ ow,     // (18,C,3,3) -> [18][576]
                    const float* __restrict__ obias,  // (18)
                    const float* __restrict__ mw,     // (9,C,3,3)  -> [9][576]
                    const float* __restrict__ mbias,  // (9)
                    int*   __restrict__ sampIdx,      // (NPIX,9,4)
                    float* __restrict__ sampW)        // (NPIX,9,4)
{
  __shared__ float sOut[4][16][28];   // 27 conv outputs per pixel (padded)

  const int wid  = threadIdx.x >> 5;
  const int lane = threadIdx.x & 31;
  const int hl   = lane >> 4;         // half: K group 0/1
  const int row  = lane & 15;         // M row / N col index
  const int tile = blockIdx.x * 4 + wid;
  const int gp0  = tile * 16;         // first global pixel of this tile
  const int b    = gp0 / HW_;
  const int p    = gp0 + row;         // this lane's A-row pixel
  const int pixb = p - b * HW_;
  const int ph   = pixb / W_;
  const int pw   = pixb - ph * W_;
  const float* xb = src + (size_t)b * C_ * HW_;

  // B column channels for this lane (combined [offset|mask] weight matrix)
  const int oc1 = 16 + row;           // second N-tile column (16..31, >=27 pad)
  const float* bp1 = (oc1 < 18) ? (ow + (size_t)oc1 * KTOT)
                   : (oc1 < 27) ? (mw + (size_t)(oc1 - 18) * KTOT)
                                : ow;                 // dummy (zeroed below)
  const float scl1 = (oc1 < 27) ? 1.f : 0.f;
  const float* bp0 = ow + (size_t)row * KTOT;         // cols 0..15 (<18)

  v8f acc0 = zero8();
  v8f acc1 = zero8();

  for (int k = 0; k < KK_; ++k) {
    const int ti = k / 3, tj = k - 3 * (k / 3);
    const int hh = ph - 1 + ti, ww = pw - 1 + tj;
    const bool vld = (hh >= 0) & (hh < H_) & (ww >= 0) & (ww < W_);
    const int hc = min(max(hh, 0), H_ - 1);
    const int wc = min(max(ww, 0), W_ - 1);
    const float* xk  = xb + hc * W_ + wc;   // + c*HW per channel
    const float* w0p = bp0 + k;             // + c*9 per channel
    const float* w1p = bp1 + k;

    #pragma unroll 4
    for (int cs = 0; cs < C_; cs += 4) {
      const int c0 = cs + 2 * hl;
      float x0 = xk[(size_t)c0 * HW_];
      float x1 = xk[(size_t)(c0 + 1) * HW_];
      v2f a;
      a.x = vld ? x0 : 0.f;
      a.y = vld ? x1 : 0.f;
      v2f b0, b1;
      b0.x = w0p[(size_t)c0 * KK_];
      b0.y = w0p[(size_t)(c0 + 1) * KK_];
      b1.x = w1p[(size_t)c0 * KK_] * scl1;
      b1.y = w1p[(size_t)(c0 + 1) * KK_] * scl1;
      acc0 = wmma4(a, b0, acc0);
      acc1 = wmma4(a, b1, acc1);
    }
  }

  // D layout: acc[r] -> (M = r + 8*hl, N = col); relay through LDS + bias
  #pragma unroll
  for (int r = 0; r < 8; ++r) {
    int m = r + 8 * hl;
    sOut[wid][m][row] = acc0[r] + obias[row];
    if (oc1 < 27) {
      float bi = (oc1 < 18) ? obias[oc1] : mbias[oc1 - 18];
      sOut[wid][m][oc1] = acc1[r] + bi;
    }
  }
  __syncthreads();

  // Bilinear sampling setup per (pixel, kernel point)
  for (int t = lane; t < 16 * KK_; t += 32) {
    int pl = t / KK_;
    int k  = t - pl * KK_;
    int gp = gp0 + pl;
    int pb2 = gp - b * HW_;
    int hh = pb2 / W_;
    int ww = pb2 - hh * W_;

    float dy = sOut[wid][pl][2 * k];
    float dx = sOut[wid][pl][2 * k + 1];
    float mk = 2.f * sigmoidf_(sOut[wid][pl][18 + k]);

    float py = (float)(hh - 1 + k / 3) + dy;
    float px = (float)(ww - 1 + k % 3) + dx;
    float y0f = floorf(py), x0f = floorf(px);
    float wy = py - y0f,    wx = px - x0f;
    int y0 = (int)y0f, x0 = (int)x0f;
    int y1 = y0 + 1,   x1 = x0 + 1;

    bool vy0 = (y0 >= 0) & (y0 <= H_ - 1);
    bool vy1 = (y1 >= 0) & (y1 <= H_ - 1);
    bool vx0 = (x0 >= 0) & (x0 <= W_ - 1);
    bool vx1 = (x1 >= 0) & (x1 <= W_ - 1);
    int y0c = min(max(y0, 0), H_ - 1), y1c = min(max(y1, 0), H_ - 1);
    int x0c = min(max(x0, 0), W_ - 1), x1c = min(max(x1, 0), W_ - 1);

    float4 wv;
    wv.x = (1.f - wy) * (1.f - wx) * mk * ((vy0 & vx0) ? 1.f : 0.f);
    wv.y = (1.f - wy) * wx         * mk * ((vy0 & vx1) ? 1.f : 0.f);
    wv.z = wy         * (1.f - wx) * mk * ((vy1 & vx0) ? 1.f : 0.f);
    wv.w = wy         * wx         * mk * ((vy1 & vx1) ? 1.f : 0.f);
    int4 iv;
    iv.x = y0c * W_ + x0c;  iv.y = y0c * W_ + x1c;
    iv.z = y1c * W_ + x0c;  iv.w = y1c * W_ + x1c;

    size_t base = ((size_t)gp * KK_ + k) * 4;
    *(int4*)  (sampIdx + base) = iv;
    *(float4*)(sampW   + base) = wv;
  }
}

// ---------------------------------------------------------------------------
// Kernel 2: deformable-conv contraction  out[b,o,p] = sum_{c,k} val * w[o,c,k]
// GEMM: M=16 pixels, N=64 channels (4 tiles), K=576 in k-major order so the
// per-(pixel,k) sampling table (LDS) is read once per k and reused for all 64
// channels. A-fragments gathered on the fly from 4 bilinear corners.
// ---------------------------------------------------------------------------
__global__ __launch_bounds__(128)
void k_deform_gemm(const float* __restrict__ src,     // (B,C,H,W)
                   const int*   __restrict__ sampIdx, // (NPIX,9,4)
                   const float* __restrict__ sampW,   // (NPIX,9,4)
                   const float* __restrict__ wt,      // (C, C*9)
                   float* __restrict__ dst)           // (B,C,H,W)
{
  __shared__ int4   sIdx[4][16 * KK_];
  __shared__ float4 sW4 [4][16 * KK_];

  const int wid  = threadIdx.x >> 5;
  const int lane = threadIdx.x & 31;
  const int hl   = lane >> 4;
  const int row  = lane & 15;
  const int tile = blockIdx.x * 4 + wid;
  const int gp0  = tile * 16;
  const int b    = gp0 / HW_;
  const float* xb = src + (size_t)b * C_ * HW_;

  for (int t = lane; t < 16 * KK_; t += 32) {
    size_t base = ((size_t)(gp0)*KK_ + t) * 4;
    sIdx[wid][t] = *(const int4*)  (sampIdx + base);
    sW4 [wid][t] = *(const float4*)(sampW   + base);
  }
  __syncthreads();

  const float* wr0 = wt + (size_t)(row)      * KTOT;
  const float* wr1 = wt + (size_t)(16 + row) * KTOT;
  const float* wr2 = wt + (size_t)(32 + row) * KTOT;
  const float* wr3 = wt + (size_t)(48 + row) * KTOT;

  v8f acc[4];
  acc[0] = zero8(); acc[1] = zero8(); acc[2] = zero8(); acc[3] = zero8();

  for (int k = 0; k < KK_; ++k) {
    const int4   iv = sIdx[wid][row * KK_ + k];  // both A elements share k
    const float4 wv = sW4 [wid][row * KK_ + k];

    #pragma unroll 4
    for (int cs = 0; cs < C_; cs += 4) {
      const int c0 = cs + 2 * hl;
      const float* p0 = xb + (size_t)c0 * HW_;
      const float* p1 = p0 + HW_;
      v2f a;
      a.x = wv.x * p0[iv.x] + wv.y * p0[iv.y] + wv.z * p0[iv.z] + wv.w * p0[iv.w];
      a.y = wv.x * p1[iv.x] + wv.y * p1[iv.y] + wv.z * p1[iv.z] + wv.w * p1[iv.w];

      const size_t kb0 = (size_t)c0 * KK_ + k;
      const size_t kb1 = kb0 + KK_;
      v2f b0 = {wr0[kb0], wr0[kb1]};
      v2f b1 = {wr1[kb0], wr1[kb1]};
      v2f b2 = {wr2[kb0], wr2[kb1]};
      v2f b3 = {wr3[kb0], wr3[kb1]};
      acc[0] = wmma4(a, b0, acc[0]);
      acc[1] = wmma4(a, b1, acc[1]);
      acc[2] = wmma4(a, b2, acc[2]);
      acc[3] = wmma4(a, b3, acc[3]);
    }
  }

  const int pixb0 = gp0 - b * HW_;
  #pragma unroll
  for (int nt = 0; nt < 4; ++nt) {
    int ch = nt * 16 + row;
    float* dp = dst + ((size_t)b * C_ + ch) * HW_ + pixb0;
    #pragma unroll
    for (int r = 0; r < 8; ++r) dp[r + 8 * hl] = acc[nt][r];
  }
}

// ---------------------------------------------------------------------------
// BatchNorm: two-pass. Stats kernel = one workgroup per channel.
// ---------------------------------------------------------------------------
__global__ __launch_bounds__(256)
void k_bn_stats(const float* __restrict__ y, float* __restrict__ stats) {
  __shared__ float ssum[256], ssq[256];
  const int c = blockIdx.x, tid = threadIdx.x;
  float s = 0.f, q = 0.f;
  for (int bb = 0; bb < B_; ++bb) {
    const float* p = y + ((size_t)bb * C_ + c) * HW_;
    for (int i = tid; i < HW_; i += 256) { float v = p[i]; s += v; q += v * v; }
  }
  ssum[tid] = s; ssq[tid] = q;
  __syncthreads();
  for (int st = 128; st > 0; st >>= 1) {
    if (tid < st) { ssum[tid] += ssum[tid + st]; ssq[tid] += ssq[tid + st]; }
    __syncthreads();
  }
  if (tid == 0) {
    float n   = (float)(B_ * HW_);
    float mu  = ssum[0] / n;
    float var = ssq[0] / n - mu * mu;
    stats[c]      = mu;
    stats[C_ + c] = rsqrtf(var + EPS_);
  }
}

__global__ __launch_bounds__(256)
void k_bn_apply(const float* __restrict__ y, const float* __restrict__ stats,
                const float* __restrict__ g, const float* __restrict__ bt,
                float* __restrict__ out, int do_relu) {
  int i = blockIdx.x * 256 + threadIdx.x;
  if (i >= B_ * C_ * HW_) return;
  int c = (i / HW_) % C_;
  float v = (y[i] - stats[c]) * stats[C_ + c] * g[c] + bt[c];
  out[i] = do_relu ? fmaxf(v, 0.f) : v;
}

// ---------------------------------------------------------------------------
// Kernel 5: both sigmoid-gate MLPs as WMMA GEMMs (K=64 then K=32, hidden via
// LDS) fused with channel blend, residual add and final ReLU.
// ---------------------------------------------------------------------------
__global__ __launch_bounds__(128)
void k_gate(const float* __restrict__ y2, const float* __restrict__ xin,
            const float* __restrict__ gw1, const float* __restrict__ gb1,
            const float* __restrict__ gw2, const float* __restrict__ gb2,
            const float* __restrict__ ow1, const float* __restrict__ ob1,
            const float* __restrict__ ow2, const float* __restrict__ ob2,
            float* __restrict__ out)
{
  __shared__ float sHg[4][16][32];
  __shared__ float sHo[4][16][32];

  const int wid  = threadIdx.x >> 5;
  const int lane = threadIdx.x & 31;
  const int hl   = lane >> 4;
  const int row  = lane & 15;
  const int tile = blockIdx.x * 4 + wid;
  const int gp0  = tile * 16;
  const int b    = gp0 / HW_;
  const int p    = gp0 + row;
  const int pb   = p - b * HW_;
  const float* yb = y2 + (size_t)b * C_ * HW_;

  // GEMM1: hidden = relu(y2 * W1^T + b1), W1 is (32,64); two MLPs in parallel
  v8f ag0 = zero8(), ag1 = zero8(), ao0 = zero8(), ao1 = zero8();
  for (int step = 0; step < 16; ++step) {
    const int kb = 4 * step + 2 * hl;
    v2f a;
    a.x = yb[(size_t)kb * HW_ + pb];
    a.y = yb[(size_t)(kb + 1) * HW_ + pb];
    const int o0 = row, o1 = 16 + row;
    v2f bg0 = {gw1[o0 * 64 + kb], gw1[o0 * 64 + kb + 1]};
    v2f bg1 = {gw1[o1 * 64 + kb], gw1[o1 * 64 + kb + 1]};
    v2f bo0 = {ow1[o0 * 64 + kb], ow1[o0 * 64 + kb + 1]};
    v2f bo1 = {ow1[o1 * 64 + kb], ow1[o1 * 64 + kb + 1]};
    ag0 = wmma4(a, bg0, ag0);
    ag1 = wmma4(a, bg1, ag1);
    ao0 = wmma4(a, bo0, ao0);
    ao1 = wmma4(a, bo1, ao1);
  }
  #pragma unroll
  for (int r = 0; r < 8; ++r) {
    int m = r + 8 * hl;
    sHg[wid][m][row]      = fmaxf(ag0[r] + gb1[row],      0.f);
    sHg[wid][m][16 + row] = fmaxf(ag1[r] + gb1[16 + row], 0.f);
    sHo[wid][m][row]      = fmaxf(ao0[r] + ob1[row],      0.f);
    sHo[wid][m][16 + row] = fmaxf(ao1[r] + ob1[16 + row], 0.f);
  }
  __syncthreads();

  // GEMM2: logits = hidden * W2^T + b2, W2 is (64,32)
  v8f g[4], o[4];
  #pragma unroll
  for (int nt = 0; nt < 4; ++nt) { g[nt] = zero8(); o[nt] = zero8(); }
  for (int step = 0; step < 8; ++step) {
    const int kb = 4 * step + 2 * hl;
    v2f aG, aO;
    aG.x = sHg[wid][row][kb]; aG.y = sHg[wid][row][kb + 1];
    aO.x = sHo[wid][row][kb]; aO.y = sHo[wid][row][kb + 1];
    #pragma unroll
    for (int nt = 0; nt < 4; ++nt) {
      int c = nt * 16 + row;
      v2f bg = {gw2[c * 32 + kb], gw2[c * 32 + kb + 1]};
      v2f bo = {ow2[c * 32 + kb], ow2[c * 32 + kb + 1]};
      g[nt] = wmma4(aG, bg, g[nt]);
      o[nt] = wmma4(aO, bo, o[nt]);
    }
  }

  // Epilogue: sigmoid gates, channel blend a[c], residual + ReLU
  const int pixb0 = gp0 - b * HW_;
  #pragma unroll
  for (int nt = 0; nt < 4; ++nt) {
    int c = nt * 16 + row;
    float ac = (c >= (C_ / 2 - 10) && c < (C_ / 2 + 10)) ? 1.2f : 1.f; // C//6=10
    size_t base = ((size_t)b * C_ + c) * HW_ + pixb0;
    #pragma unroll
    for (int r = 0; r < 8; ++r) {
      int m = r + 8 * hl;
      float ga = sigmoidf_(g[nt][r] + gb2[c]);
      float oa = sigmoidf_(o[nt][r] + ob2[c]);
      float v  = y2[base + m];
      float res = v * ((ga * ac + oa * (2.f - ac)) * 0.5f) + xin[base + m];
      out[base + m] = fmaxf(res, 0.f);
    }
  }
}

// ---------------------------------------------------------------------------
extern "C" void kernel_launch(void* const* d_in, const int* in_sizes, int n_in,
                              void* d_out, int out_size, void* d_ws, size_t ws_size,
                              hipStream_t stream) {
  (void)in_sizes; (void)n_in; (void)out_size; (void)ws_size;
  const float* x   = (const float*)d_in[0];
  const float* o1w = (const float*)d_in[1];
  const float* o1b = (const float*)d_in[2];
  const float* m1w = (const float*)d_in[3];
  const float* m1b = (const float*)d_in[4];
  const float* w1  = (const float*)d_in[5];
  const float* g1  = (const float*)d_in[6];
  const float* b1  = (const float*)d_in[7];
  const float* o2w = (const float*)d_in[8];
  const float* o2b = (const float*)d_in[9];
  const float* m2w = (const float*)d_in[10];
  const float* m2b = (const float*)d_in[11];
  const float* w2  = (const float*)d_in[12];
  const float* g2  = (const float*)d_in[13];
  const float* b2  = (const float*)d_in[14];
  const float* gw1 = (const float*)d_in[15];
  const float* gb1 = (const float*)d_in[16];
  const float* gw2 = (const float*)d_in[17];
  const float* gb2 = (const float*)d_in[18];
  const float* ow1 = (const float*)d_in[19];
  const float* ob1 = (const float*)d_in[20];
  const float* ow2 = (const float*)d_in[21];
  const float* ob2 = (const float*)d_in[22];
  float* out = (float*)d_out;

  char* ws = (char*)d_ws;
  size_t off = 0;
  auto alloc = [&](size_t bytes) -> void* {
    void* pp = ws + off;
    off = (off + bytes + 255) & ~(size_t)255;
    return pp;
  };
  int*   sampIdx = (int*)  alloc((size_t)NPIX * KK_ * 4 * sizeof(int));
  float* sampW   = (float*)alloc((size_t)NPIX * KK_ * 4 * sizeof(float));
  float* yA      = (float*)alloc((size_t)B_ * C_ * HW_ * sizeof(float));
  float* yB      = (float*)alloc((size_t)B_ * C_ * HW_ * sizeof(float));
  float* stats   = (float*)alloc(2 * C_ * sizeof(float));

  const dim3 blk(128);
  const dim3 grd(NPIX / 64);              // 4 waves/block * 16 px/wave
  const int  nElem   = B_ * C_ * HW_;
  const dim3 grdElem((nElem + 255) / 256);

  // Stage 1: deform_conv -> BN -> ReLU
  k_sample_setup<<<grd, blk, 0, stream>>>(x, o1w, o1b, m1w, m1b, sampIdx, sampW);
  k_deform_gemm <<<grd, blk, 0, stream>>>(x, sampIdx, sampW, w1, yA);
  k_bn_stats    <<<C_, 256, 0, stream>>>(yA, stats);
  k_bn_apply    <<<grdElem, 256, 0, stream>>>(yA, stats, g1, b1, yB, 1);

  // Stage 2: deform_conv -> BN
  k_sample_setup<<<grd, blk, 0, stream>>>(yB, o2w, o2b, m2w, m2b, sampIdx, sampW);
  k_deform_gemm <<<grd, blk, 0, stream>>>(yB, sampIdx, sampW, w2, yA);
  k_bn_stats    <<<C_, 256, 0, stream>>>(yA, stats);
  k_bn_apply    <<<grdElem, 256, 0, stream>>>(yA, stats, g2, b2, yB, 0);

  // Gates + residual + ReLU
  k_gate<<<grd, blk, 0, stream>>>(yB, x, gw1, gb1, gw2, gb2,
                                  ow1, ob1, ow2, ob2, out);
}